// pixel_shuffle_layer_70858370449951
// MI455X (gfx1250) — compile-verified
//
#include <hip/hip_runtime.h>

// Space-to-depth (s=2) for input (B=4, C=3, T=16, H=512, W=512) fp32.
// out[b][k][t][hc][wc] = in[b][c][t][2*hc+sh][2*wc+sw],  k = sh*6 + sw*3 + c.
//
// Roofline: 0 FLOPs, 402.6 MB total traffic -> ~17.3 us floor at 23.3 TB/s.
// Strategy: per-block scalar (SALU) index decomposition, per-thread constant
// base pointers, fully unrolled x4 body with immediate-offset vmem only:
//   4x global_load_b128 (NT) + 8x global_store_b64 (NT), zero loop VALU.
//
// Block covers 1024 consecutive float4 groups of the input:
//   blockIdx = (b*3+c)*1024 + r,  r = t*64 + hb   (t in [0,16), hb in [0,64))
//   thread handles local = tid + 256*i, i = 0..3:
//     w4 = tid & 127        (iteration-invariant)
//     sh = tid >> 7         (iteration-invariant)
//     h  = hb*8 + sh + 2*i  -> hc = hb*4 + i, advances by +1 per iteration

typedef __attribute__((ext_vector_type(4))) float f32x4;
typedef __attribute__((ext_vector_type(2))) float f32x2;

#define K_STRIDE   1048576u   // 16*256*256  (k-plane stride in floats)
#define SW_OFF     3145728u   // 3 * K_STRIDE (sw=0 -> sw=1 plane distance)
#define SH_OFF     6291456u   // 6 * K_STRIDE (sh contribution)

__global__ __launch_bounds__(256) void
space_to_depth_s2_kernel(const f32x4* __restrict__ in, float* __restrict__ out)
{
    // ---- block-uniform (SALU) ----
    const unsigned ct = blockIdx.x >> 10;      // b*3 + c, 0..11
    const unsigned r  = blockIdx.x & 1023u;
    const unsigned c  = ct % 3u;               // scalar mod-3, once per block
    const unsigned b  = ct / 3u;
    const unsigned t  = r >> 6;
    const unsigned hb = r & 63u;

    // input base (float4 units) for this block
    const f32x4* inp = in + ((ct << 20) + (r << 10)) + threadIdx.x;

    // output base (floats): (b*12+c)*K + t*65536 + hb*1024
    const unsigned out_base =
        (b * 12u + c) * K_STRIDE + (t << 16) + (hb << 10);

    // ---- thread-invariant (VALU, once) ----
    const unsigned w4 = threadIdx.x & 127u;
    const unsigned sh = threadIdx.x >> 7;      // 0 or 1
    float* o0 = out + out_base + sh * SH_OFF + (w4 << 1);  // sw = 0 row
    float* o1 = o0 + SW_OFF;                               // sw = 1 row

    // ---- pure vmem body: immediate offsets only ----
#pragma unroll
    for (int i = 0; i < 4; ++i) {
        const f32x4 v = __builtin_nontemporal_load(inp + i * 256);
        const f32x2 even = { v.x, v.z };   // w parities 0 -> row k
        const f32x2 odd  = { v.y, v.w };   // w parities 1 -> row k+3
        __builtin_nontemporal_store(even, (f32x2*)(o0 + i * 256));
        __builtin_nontemporal_store(odd,  (f32x2*)(o1 + i * 256));
    }
}

extern "C" void kernel_launch(void* const* d_in, const int* in_sizes, int n_in,
                              void* d_out, int out_size, void* d_ws, size_t ws_size,
                              hipStream_t stream)
{
    const f32x4* in  = (const f32x4*)d_in[0];
    float*       out = (float*)d_out;

    const unsigned total4 = (unsigned)(in_sizes[0] / 4);  // float4 groups
    unsigned blocks = total4 >> 10;                       // 1024 groups/block
    if (blocks == 0) blocks = 1;                          // (fixed-shape harness)

    space_to_depth_s2_kernel<<<blocks, 256, 0, stream>>>(in, out);
}